// JetECF_42717744726558
// MI455X (gfx1250) — compile-verified
//
#include <hip/hip_runtime.h>
#include <hip/hip_bf16.h>

// Jet ECF (beta=2) via rank-4 factorization of the R^2 matrix:
//   R2_ij = u_i . v_j,  u=(r,1,-2eta,-2phi), v=(1,r,eta,phi), r=eta^2+phi^2
//   G = sum_i pt_i * v_i u_i^T   (4x4)
//   ecf1 = G[0][1];  ecf2 = 0.5 * sum_l G[0][l]*G[l][1];  ecf3 = tr(G^3)/6
// G accumulated on the fp32 WMMA pipe (V_WMMA_F32_16X16X4_F32): each wmma
// consumes 4 particles (rank-1 updates into a 16x16 f32 acc whose top-left
// 4x4 block is G). Fragments are pre-staged in LDS in WMMA layout so the
// inner loop is 2x ds_load_b64 + 1x v_wmma, branch-free.

typedef __attribute__((ext_vector_type(2))) float v2f;
typedef __attribute__((ext_vector_type(8))) float v8f;

#define NPART 200
#define NPAD  208           // pad to multiple of 4 (K of the wmma)
#define NPAIR (NPAD / 2)    // 104 particle pairs
#define NCHUNK (NPAD / 4)   // 52 chunks of 4 particles (1 wmma each)
#define NWAVES 8            // 256 threads / wave32

__global__ __launch_bounds__(256) void jet_ecf_kernel(const float* __restrict__ x,
                                                      float* __restrict__ out,
                                                      int nbatch) {
    __shared__ float s_pt[NPAD];
    __shared__ float s_eta[NPAD];
    __shared__ float s_phi[NPAD];
    __shared__ float s_r[NPAD];
    // WMMA-ready fragments: s_a[pair*16 + m] = {v_{2p}[m], v_{2p+1}[m]}
    //                       s_b[pair*16 + n] = {z_{2p}[n], z_{2p+1}[n]}, z = pt*u
    __shared__ v2f s_a[NPAIR * 16];
    __shared__ v2f s_b[NPAIR * 16];
    __shared__ float s_red[NWAVES][16];  // per-wave partial G (4x4)

    const int b = blockIdx.x;
    if (b >= nbatch) return;
    const int t = threadIdx.x;
    const float* xb = x + (size_t)b * (NPART * 3);

    // Stage particle data into LDS (zero the padding region first).
    for (int i = t; i < NPAD; i += 256) {
        s_pt[i] = 0.0f; s_eta[i] = 0.0f; s_phi[i] = 0.0f;
    }
    __syncthreads();
    for (int i = t; i < NPART; i += 256) {
        s_pt[i]  = xb[i * 3 + 0];
        s_eta[i] = xb[i * 3 + 1];
        s_phi[i] = xb[i * 3 + 2];
    }
    __syncthreads();
    for (int i = t; i < NPAD; i += 256) {
        const float e = s_eta[i], f = s_phi[i];
        s_r[i] = e * e + f * f;
    }
    __syncthreads();

    // Build WMMA fragments in LDS. All ternary arms are registers -> cndmask.
    for (int j = t; j < NPAIR * 16; j += 256) {
        const int pair = j >> 4;
        const int m    = j & 15;
        const int p0 = pair * 2, p1 = p0 + 1;
        const float pt0 = s_pt[p0],  pt1 = s_pt[p1];
        const float e0  = s_eta[p0], e1  = s_eta[p1];
        const float f0  = s_phi[p0], f1  = s_phi[p1];
        const float r0  = s_r[p0],   r1  = s_r[p1];

        const float v0 = (m == 0) ? 1.0f : (m == 1) ? r0 : (m == 2) ? e0 : (m == 3) ? f0 : 0.0f;
        const float v1 = (m == 0) ? 1.0f : (m == 1) ? r1 : (m == 2) ? e1 : (m == 3) ? f1 : 0.0f;
        const float u0 = (m == 0) ? r0 : (m == 1) ? 1.0f
                        : (m == 2) ? -2.0f * e0 : (m == 3) ? -2.0f * f0 : 0.0f;
        const float u1 = (m == 0) ? r1 : (m == 1) ? 1.0f
                        : (m == 2) ? -2.0f * e1 : (m == 3) ? -2.0f * f1 : 0.0f;

        v2f va; va[0] = v0;        va[1] = v1;
        v2f vb; vb[0] = pt0 * u0;  vb[1] = pt1 * u1;
        s_a[j] = va;
        s_b[j] = vb;
    }
    __syncthreads();

    const int wave = t >> 5;
    const int lane = t & 31;
    const int idx  = lane & 15;  // row m (A) / col n (B)
    const int hx   = lane >> 4;  // half-wave = which K-pair of the chunk

    v8f acc = {};  // 16x16 f32 accumulator; rows 0..3 x cols 0..3 == G

    // Fragment address for chunk c: (c*2 + hx)*16 + idx == c*32 + lane.
    // Inner loop: 2x ds_load_b64 + 1x v_wmma, no divergence (EXEC all-ones).
    for (int c = wave; c < NCHUNK; c += NWAVES) {
        const int a = c * 32 + lane;
        const v2f A = s_a[a];
        const v2f B = s_b[a];
        acc = __builtin_amdgcn_wmma_f32_16x16x4_f32(
            /*neg_a=*/false, A, /*neg_b=*/false, B,
            /*c_mod=*/(short)0, acc, /*reuse_a=*/false, /*reuse_b=*/false);
    }

    // D layout: VGPR v holds row M=v for lanes 0-15, col N=lane.
    // Need rows 0..3, cols 0..3.
    if (hx == 0 && idx < 4) {
#pragma unroll
        for (int v = 0; v < 4; ++v) s_red[wave][v * 4 + idx] = acc[v];
    }
    __syncthreads();

    if (t == 0) {
        float G[4][4];
#pragma unroll
        for (int m = 0; m < 4; ++m)
#pragma unroll
            for (int n = 0; n < 4; ++n) {
                float s = 0.0f;
#pragma unroll
                for (int w = 0; w < NWAVES; ++w) s += s_red[w][m * 4 + n];
                G[m][n] = s;
            }

        const float ecf1 = G[0][1];  // sum(pt)
        const float ecf2 = 0.5f * (G[0][0] * G[0][1] + G[0][1] * G[1][1] +
                                   G[0][2] * G[2][1] + G[0][3] * G[3][1]);

        float G2[4][4];
#pragma unroll
        for (int i = 0; i < 4; ++i)
#pragma unroll
            for (int j = 0; j < 4; ++j) {
                float s = 0.0f;
#pragma unroll
                for (int k = 0; k < 4; ++k) s += G[i][k] * G[k][j];
                G2[i][j] = s;
            }
        float tr3 = 0.0f;
#pragma unroll
        for (int i = 0; i < 4; ++i)
#pragma unroll
            for (int j = 0; j < 4; ++j) tr3 += G2[i][j] * G[j][i];
        const float ecf3 = tr3 * (1.0f / 6.0f);

        const float e13 = ecf1 * ecf1 * ecf1;
        const float e23 = ecf2 * ecf2 * ecf2;
        const float d2  = ecf3 * e13 / (e23 + 1e-7f);

        float* o = out + (size_t)b * 4;
        o[0] = ecf1;
        o[1] = ecf2;
        o[2] = ecf3;
        o[3] = d2;
    }
}

extern "C" void kernel_launch(void* const* d_in, const int* in_sizes, int n_in,
                              void* d_out, int out_size, void* d_ws, size_t ws_size,
                              hipStream_t stream) {
    (void)n_in; (void)d_ws; (void)ws_size; (void)out_size;
    const float* x = (const float*)d_in[0];
    float* out = (float*)d_out;
    const int nbatch = in_sizes[0] / (NPART * 3);  // 2048
    jet_ecf_kernel<<<nbatch, 256, 0, stream>>>(x, out, nbatch);
}